// FirstOrderLinear_68032281969207
// MI455X (gfx1250) — compile-verified
//
#include <hip/hip_runtime.h>

// FirstOrderLinear scan: s_{t+1} = s_t + a*(x_t - s_t), a = dt/max(tau,dt),
// out_t = s_{t+1} + bias.  T=8192, N=4096, fp32.  Memory-bound.
// Chunked affine-scan decomposition + async global->LDS streaming pipeline.
// Pass-3 output stores are non-temporal so the 128 MiB input stays resident
// in the 192 MiB L2 between pass 1 and pass 3.

typedef float v4f __attribute__((ext_vector_type(4)));

namespace {
constexpr int   kT      = 8192;
constexpr int   kN      = 4096;
constexpr int   kNCol   = kN / 4;         // 1024 float4 columns
constexpr int   kChunks = 64;             // K
constexpr int   kRows   = kT / kChunks;   // L = 128 rows per chunk
constexpr int   kBC     = 128;            // float4 columns (threads) per block
constexpr int   kDepth  = 16;             // async pipeline depth (power of two)
constexpr float kDt     = 0.01f;
static_assert(kRows == 128, "pass2 uses 7 squarings for r^128");
static_assert((kDepth & (kDepth - 1)) == 0 && kDepth <= kRows, "depth");
}

// ---- CDNA5 async global->LDS copy (128b per lane), ASYNCcnt-tracked ----
__device__ __forceinline__ void async_copy_f4(unsigned lds_addr,
                                              const float* sbase,
                                              unsigned voff) {
  // GVS mode: mem = SADDR(64b sgpr pair) + VADDR(32b per-lane) ; dest = LDS byte addr
  asm volatile("global_load_async_to_lds_b128 %0, %1, %2"
               :: "v"(lds_addr), "v"(voff), "s"(sbase)
               : "memory");
}
__device__ __forceinline__ void wait_async_le_depth_m1() {
  asm volatile("s_wait_asynccnt 0xf" ::: "memory");   // <= kDepth-1 outstanding
}
__device__ __forceinline__ void wait_ds0() {
  asm volatile("s_wait_dscnt 0x0" ::: "memory");      // LDS reads retired
}

// ---- Pass 1: per-chunk carry B (state after L rows starting from s=0) ----
__global__ __launch_bounds__(kBC) void foli_pass1_carry(
    const float* __restrict__ inp, const float* __restrict__ tau,
    float* __restrict__ carryB) {
  __shared__ float4 buf[kDepth][kBC];
  const int k    = blockIdx.x;
  const int gcol = blockIdx.y * kBC + threadIdx.x;

  const float4 t4 = reinterpret_cast<const float4*>(tau)[gcol];
  const float ax = kDt / fmaxf(t4.x, kDt);
  const float ay = kDt / fmaxf(t4.y, kDt);
  const float az = kDt / fmaxf(t4.z, kDt);
  const float aw = kDt / fmaxf(t4.w, kDt);

  const float*   base = inp + (size_t)k * kRows * kN;
  const unsigned lds0 = (unsigned)(size_t)(&buf[0][threadIdx.x]);
  const unsigned colb = (unsigned)gcol * 16u;

  #pragma unroll
  for (int j = 0; j < kDepth; ++j)
    async_copy_f4(lds0 + (unsigned)j * (kBC * 16), base,
                  (unsigned)j * (unsigned)(kN * 4) + colb);

  float sx = 0.f, sy = 0.f, sz = 0.f, sw = 0.f;
  for (int j = 0; j < kRows; ++j) {
    wait_async_le_depth_m1();                 // row j landed in LDS
    const int slot = j & (kDepth - 1);
    const float4 x = buf[slot][threadIdx.x];  // ds_load_b128
    wait_ds0();                               // slot free for reuse
    const int jn = j + kDepth;
    if (jn < kRows)
      async_copy_f4(lds0 + (unsigned)slot * (kBC * 16), base,
                    (unsigned)jn * (unsigned)(kN * 4) + colb);
    sx = fmaf(ax, x.x - sx, sx);
    sy = fmaf(ay, x.y - sy, sy);
    sz = fmaf(az, x.z - sz, sz);
    sw = fmaf(aw, x.w - sw, sw);
  }
  reinterpret_cast<float4*>(carryB)[(size_t)k * kNCol + gcol] =
      make_float4(sx, sy, sz, sw);
}

// ---- Pass 2: serial combine of chunk carries -> chunk start states ----
__global__ __launch_bounds__(256) void foli_pass2_scan(
    const float* __restrict__ inp, const float* __restrict__ tau,
    const float* __restrict__ carryB, float* __restrict__ startS) {
  const int col = blockIdx.x * blockDim.x + threadIdx.x;
  if (col >= kNCol) return;
  const float4 t4 = reinterpret_cast<const float4*>(tau)[col];
  float rx = 1.f - kDt / fmaxf(t4.x, kDt);
  float ry = 1.f - kDt / fmaxf(t4.y, kDt);
  float rz = 1.f - kDt / fmaxf(t4.z, kDt);
  float rw = 1.f - kDt / fmaxf(t4.w, kDt);
  #pragma unroll
  for (int i = 0; i < 7; ++i) { rx *= rx; ry *= ry; rz *= rz; rw *= rw; } // r^128

  float4 s = reinterpret_cast<const float4*>(inp)[col];  // s_0 = input[0]
  #pragma unroll 8
  for (int k = 0; k < kChunks; ++k) {
    reinterpret_cast<float4*>(startS)[(size_t)k * kNCol + col] = s;
    const float4 B = reinterpret_cast<const float4*>(carryB)[(size_t)k * kNCol + col];
    s.x = fmaf(rx, s.x, B.x);
    s.y = fmaf(ry, s.y, B.y);
    s.z = fmaf(rz, s.z, B.z);
    s.w = fmaf(rw, s.w, B.w);
  }
}

// ---- Pass 3: re-scan each chunk from its correct start state, emit out ----
__global__ __launch_bounds__(kBC) void foli_pass3_emit(
    const float* __restrict__ inp, const float* __restrict__ tau,
    const float* __restrict__ bias, const float* __restrict__ startS,
    float* __restrict__ out) {
  __shared__ float4 buf[kDepth][kBC];
  const int k    = blockIdx.x;
  const int gcol = blockIdx.y * kBC + threadIdx.x;

  const float4 t4 = reinterpret_cast<const float4*>(tau)[gcol];
  const float ax = kDt / fmaxf(t4.x, kDt);
  const float ay = kDt / fmaxf(t4.y, kDt);
  const float az = kDt / fmaxf(t4.z, kDt);
  const float aw = kDt / fmaxf(t4.w, kDt);
  const float4 b4 = reinterpret_cast<const float4*>(bias)[gcol];
  const float4 s4 =
      reinterpret_cast<const float4*>(startS)[(size_t)k * kNCol + gcol];
  float sx = s4.x, sy = s4.y, sz = s4.z, sw = s4.w;

  const float*   base = inp + (size_t)k * kRows * kN;
  const unsigned lds0 = (unsigned)(size_t)(&buf[0][threadIdx.x]);
  const unsigned colb = (unsigned)gcol * 16u;
  v4f* outb = reinterpret_cast<v4f*>(out) + (size_t)k * kRows * kNCol + gcol;

  #pragma unroll
  for (int j = 0; j < kDepth; ++j)
    async_copy_f4(lds0 + (unsigned)j * (kBC * 16), base,
                  (unsigned)j * (unsigned)(kN * 4) + colb);

  for (int j = 0; j < kRows; ++j) {
    wait_async_le_depth_m1();
    const int slot = j & (kDepth - 1);
    const float4 x = buf[slot][threadIdx.x];
    wait_ds0();
    const int jn = j + kDepth;
    if (jn < kRows)
      async_copy_f4(lds0 + (unsigned)slot * (kBC * 16), base,
                    (unsigned)jn * (unsigned)(kN * 4) + colb);
    sx = fmaf(ax, x.x - sx, sx);
    sy = fmaf(ay, x.y - sy, sy);
    sz = fmaf(az, x.z - sz, sz);
    sw = fmaf(aw, x.w - sw, sw);
    // Non-temporal store: write-once stream; keep the input resident in L2.
    v4f o;
    o.x = sx + b4.x; o.y = sy + b4.y; o.z = sz + b4.z; o.w = sw + b4.w;
    __builtin_nontemporal_store(o, &outb[(size_t)j * kNCol]);
  }
}

extern "C" void kernel_launch(void* const* d_in, const int* in_sizes, int n_in,
                              void* d_out, int out_size, void* d_ws, size_t ws_size,
                              hipStream_t stream) {
  const float* inp  = (const float*)d_in[0];   // [T, N]
  const float* tau  = (const float*)d_in[1];   // [N]
  const float* bias = (const float*)d_in[2];   // [N]
  float* out = (float*)d_out;                  // [T*N]

  float* carryB = (float*)d_ws;                         // [K, N]
  float* startS = carryB + (size_t)kChunks * kN;        // [K, N]

  const dim3 grid(kChunks, kNCol / kBC);                // (64, 8)
  foli_pass1_carry<<<grid, kBC, 0, stream>>>(inp, tau, carryB);
  foli_pass2_scan<<<dim3(kNCol / 256), 256, 0, stream>>>(inp, tau, carryB, startS);
  foli_pass3_emit<<<grid, kBC, 0, stream>>>(inp, tau, bias, startS, out);
}